// RecursiveNet_69088843924181
// MI455X (gfx1250) — compile-verified
//
#include <hip/hip_runtime.h>

typedef float v2f __attribute__((ext_vector_type(2)));
typedef float v4f __attribute__((ext_vector_type(4)));
typedef float v8f __attribute__((ext_vector_type(8)));

#define XS 132          // padded LDS stride (floats) per X column: 528 B = 33*16 B
#define YS 132          // padded LDS stride per Y position
#define XCOLS 33        // 32 conv positions need 33 input columns
#define SLOPE 0.2f

// ---------------------------------------------------------------------------
// Precompute per-level constant vector:
//   c[i][o] = b[o] + sum_k (W[o][128+k][0] + W[o][128+k][1]) * depth[i][k]
// (the broadcast-embedding half of the conv collapses to this constant)
// ---------------------------------------------------------------------------
__global__ __launch_bounds__(128) void rn_cvec_kernel(const float* __restrict__ W,
                                                      const float* __restrict__ b,
                                                      const float* __restrict__ depth,
                                                      float* __restrict__ cbuf) {
  const int i = blockIdx.x;    // level 0..17
  const int o = threadIdx.x;   // out channel 0..127
  const float* wrow = W + (long)o * 512 + 256;   // W[o][128+k][kk] at wrow[2k+kk]
  const float* drow = depth + (long)i * 128;
  float s = b[o];
#pragma unroll 4
  for (int k = 0; k < 128; ++k)
    s += (wrow[2 * k] + wrow[2 * k + 1]) * drow[k];
  cbuf[i * 128 + o] = s;
}

// ---------------------------------------------------------------------------
// One pyramid level: y[:,p] = leakyrelu(A x[:,p] + B x[:,p+1] + c),
// then x_out[:,q] = max(y[:,2q], y[:,2q+1]).
// Block: 256 threads = 8 waves; wave w owns out-channel rows [16w, 16w+16).
// Each block tile covers 32 conv positions -> 16 pooled outputs.
// ---------------------------------------------------------------------------
__global__ __launch_bounds__(256) void rn_level_kernel(
    const float* __restrict__ in, const int* __restrict__ perm, int use_perm,
    const float* __restrict__ W, const float* __restrict__ cvec,
    float* __restrict__ out, int L_in) {
  __shared__ float xs[XCOLS * XS];   // X tile: [col][ch]
  __shared__ float ys[32 * YS];      // Y tile: [pos][ch]

  const int tid = threadIdx.x;
  const int lane = tid & 31;
  const int wv = tid >> 5;
  const int lane15 = lane & 15;
  const int half = lane >> 4;        // 0: lanes 0-15, 1: lanes 16-31
  const int ks = half * 2;           // K offset inside a group of 4 (ISA A/B layout)
  const int m0 = wv * 16;            // this wave's out-channel block
  const int row = m0 + lane15;

  const long Lo = (long)L_in - 1;    // conv output length
  const long Lp = Lo >> 1;           // pooled output length
  const long tiles = (Lo + 31) >> 5;

  // Resident weight fragments. ISA 16x4 f32 A layout:
  //   lanes 0-15: v0=K0,v1=K1 ; lanes 16-31: v0=K2,v1=K3 (M = lane%16)
  v2f wA[32], wB[32];
  const float* wr = W + (long)row * 512;   // W[row][k][kk] at wr[2k+kk]
#pragma unroll
  for (int t = 0; t < 32; ++t) {
    const int k0 = 4 * t + ks;
    v2f a, bb;
    a[0]  = wr[2 * k0];         a[1]  = wr[2 * (k0 + 1)];        // kernel tap 0 -> A
    bb[0] = wr[2 * k0 + 1];     bb[1] = wr[2 * (k0 + 1) + 1];    // kernel tap 1 -> B
    wA[t] = a; wB[t] = bb;
  }
  // C/D layout: VGPR r holds M = m0+r (lanes 0-15) / m0+r+8 (lanes 16-31)
  float cinit[8];
#pragma unroll
  for (int r = 0; r < 8; ++r) cinit[r] = cvec[m0 + r + half * 8];

  for (long tk = blockIdx.x; tk < tiles; tk += gridDim.x) {
    const long p0 = tk << 5;

    // Stage 33 input columns (128 ch each) into LDS as 16B vectors; level 0
    // gathers whole 512B rows via perm (32 consecutive lanes = one burst).
    for (int idx = tid; idx < XCOLS * 32; idx += 256) {
      const int col = idx >> 5, ch = (idx & 31) << 2;
      const long gp = p0 + col;
      v4f v = {0.0f, 0.0f, 0.0f, 0.0f};
      if (gp < (long)L_in) {
        const long r2 = use_perm ? (long)perm[gp] : gp;
        v = *(const v4f*)(in + r2 * 128 + ch);
      }
      *(v4f*)&xs[col * XS + ch] = v;
    }

    // Prefetch the tile this block processes next (one full tile-compute
    // early): pull its gathered 128B lines toward L2/WGP.
    {
      const long tk2 = tk + gridDim.x;
      if (tk2 < tiles) {
        const int idx = tid;                   // 33 cols * 4 cachelines = 132 <= 256
        if (idx < XCOLS * 4) {
          const int col = idx >> 2, seg = (idx & 3) << 5;   // 32 floats per 128B line
          const long gp = (tk2 << 5) + col;
          if (gp < (long)L_in) {
            const long r2 = use_perm ? (long)perm[gp] : gp;
            __builtin_prefetch(in + r2 * 128 + seg, 0, 3);  // -> global_prefetch_b8
          }
        }
      }
    }
    __syncthreads();

    v8f acc0, acc1;
#pragma unroll
    for (int r = 0; r < 8; ++r) { acc0[r] = cinit[r]; acc1[r] = cinit[r]; }

    // B-operand (4x16 X chunk): lane reads K-adjacent float2 from its column
    // -> single 8B-aligned ds_load_b64 per WMMA operand.
    const v2f* xA0 = (const v2f*)&xs[(lane15 + 0)  * XS + ks];
    const v2f* xB0 = (const v2f*)&xs[(lane15 + 1)  * XS + ks];
    const v2f* xA1 = (const v2f*)&xs[(lane15 + 16) * XS + ks];
    const v2f* xB1 = (const v2f*)&xs[(lane15 + 17) * XS + ks];

#pragma unroll
    for (int t = 0; t < 32; ++t) {   // K = 128 in chunks of 4; alternate the
      // two independent accumulator chains so they can interleave in the XDL.
      acc0 = __builtin_amdgcn_wmma_f32_16x16x4_f32(false, wA[t], false, xA0[2 * t],
                                                   (short)0, acc0, false, false);
      acc1 = __builtin_amdgcn_wmma_f32_16x16x4_f32(false, wA[t], false, xA1[2 * t],
                                                   (short)0, acc1, false, false);
      acc0 = __builtin_amdgcn_wmma_f32_16x16x4_f32(false, wB[t], false, xB0[2 * t],
                                                   (short)0, acc0, false, false);
      acc1 = __builtin_amdgcn_wmma_f32_16x16x4_f32(false, wB[t], false, xB1[2 * t],
                                                   (short)0, acc1, false, false);
    }

    // leaky_relu + stage Y in LDS (pos = N, ch = M)
#pragma unroll
    for (int r = 0; r < 8; ++r) {
      float v = acc0[r]; v = (v >= 0.0f) ? v : SLOPE * v;
      ys[(lane15 + 0)  * YS + m0 + r + half * 8] = v;
      float u = acc1[r]; u = (u >= 0.0f) ? u : SLOPE * u;
      ys[(lane15 + 16) * YS + m0 + r + half * 8] = u;
    }
    __syncthreads();

    // MaxPool(2,2): 16 pooled outputs per tile, floor semantics via Lp guard.
    // Vectorized: b128 LDS reads + b128 global stores.
    const long q0 = tk << 4;
    for (int idx = tid; idx < 16 * 32; idx += 256) {
      const int q = idx >> 5, ch = (idx & 31) << 2;
      if (q0 + q < Lp) {
        v4f a = *(const v4f*)&ys[(2 * q) * YS + ch];
        v4f c = *(const v4f*)&ys[(2 * q + 1) * YS + ch];
        v4f m;
#pragma unroll
        for (int r = 0; r < 4; ++r) m[r] = fmaxf(a[r], c[r]);
        *(v4f*)(out + (q0 + q) * 128 + ch) = m;
      }
    }
    __syncthreads();   // protect xs/ys before next tile
  }
}

// ---------------------------------------------------------------------------
extern "C" void kernel_launch(void* const* d_in, const int* in_sizes, int n_in,
                              void* d_out, int out_size, void* d_ws, size_t ws_size,
                              hipStream_t stream) {
  (void)in_sizes; (void)n_in; (void)out_size; (void)ws_size;
  const float* x     = (const float*)d_in[0];   // [524288,128]
  const float* depth = (const float*)d_in[1];   // [32,128]
  const float* W     = (const float*)d_in[2];   // [128,256,2]
  const float* b     = (const float*)d_in[3];   // [128]
  const int*   perm  = (const int*)d_in[4];     // [524288]
  float* outp = (float*)d_out;                  // [1,128]

  char* ws = (char*)d_ws;
  float* cbuf = (float*)ws;                                   // 18*128 floats
  float* buf0 = (float*)(ws + 16384);                         // 262144*512 B
  float* buf1 = (float*)(ws + 16384 + 262144L * 512);         // 131072*512 B

  rn_cvec_kernel<<<18, 128, 0, stream>>>(W, b, depth, cbuf);

  long L = 524288;
  const float* cur = x;
  int use_perm = 1;
  for (int lvl = 0; lvl < 18; ++lvl) {
    const long Lo = L - 1;
    const long Lp = Lo >> 1;
    float* o = (Lp == 1) ? outp : ((lvl & 1) ? buf1 : buf0);
    const long tiles = (Lo + 31) >> 5;
    const long gl = tiles < 2048 ? tiles : 2048;
    rn_level_kernel<<<dim3((unsigned)gl), dim3(256), 0, stream>>>(
        cur, perm, use_perm, W, cbuf + lvl * 128, o, (int)L);
    cur = o;
    use_perm = 0;
    L = Lp;
  }
}